// TensorIntegration_54674933678524
// MI455X (gfx1250) — compile-verified
//
#include <hip/hip_runtime.h>

typedef __attribute__((ext_vector_type(2))) float v2f;
typedef __attribute__((ext_vector_type(8))) float v8f;

__device__ __forceinline__ int degi(int x){ return x>=9 ? 3 : (x>=4 ? 2 : (x>=1 ? 1 : 0)); }
__device__ __forceinline__ int iabs(int x){ return x < 0 ? -x : x; }

// ---------------------------------------------------------------------------
// CG table init: real-basis Clebsch-Gordan tensor for L=3, shape (16,16,16).
// Recomputed deterministically every launch into d_ws (16 KB).
// ---------------------------------------------------------------------------
__device__ double dfact(int n){ double r = 1.0; for (int t = 2; t <= n; ++t) r *= (double)t; return r; }

__device__ double cg_cplx(int j1,int j2,int j3,int m1,int m2,int m3){
  if (m3 != m1 + m2) return 0.0;
  if (j3 < iabs(j1 - j2) || j3 > j1 + j2) return 0.0;
  double pre = sqrt((2.0*j3 + 1.0) * dfact(j3+j1-j2) * dfact(j3-j1+j2) * dfact(j1+j2-j3) / dfact(j1+j2+j3+1));
  pre *= sqrt(dfact(j3+m3)*dfact(j3-m3)*dfact(j1-m1)*dfact(j1+m1)*dfact(j2-m2)*dfact(j2+m2));
  int kmin = max(0, max(j2 - j3 - m1, j1 - j3 + m2));
  int kmax = min(j1 + j2 - j3, min(j1 - m1, j2 + m2));
  double s = 0.0;
  for (int k = kmin; k <= kmax; ++k){
    double d = dfact(k)*dfact(j1+j2-j3-k)*dfact(j1-m1-k)*dfact(j2+m2-k)*dfact(j3-j2+m1+k)*dfact(j3-j1-m2+k);
    s += ((k & 1) ? -1.0 : 1.0) / d;
  }
  return pre * s;
}

struct UT { int m; double re, im; };

__device__ int urow(int l, int mr, UT* t){
  const double s2 = 0.70710678118654752440;
  if (mr == 0){ t[0].m = 0; t[0].re = 1.0; t[0].im = 0.0; return 1; }
  if (mr > 0){
    int m = mr; double sg = (m & 1) ? -1.0 : 1.0;
    t[0].m =  m; t[0].re = sg * s2; t[0].im = 0.0;
    t[1].m = -m; t[1].re = s2;      t[1].im = 0.0;
    return 2;
  }
  int m = -mr; double sg = (m & 1) ? -1.0 : 1.0;
  t[0].m =  m; t[0].re = 0.0; t[0].im = -sg * s2;
  t[1].m = -m; t[1].re = 0.0; t[1].im = s2;
  return 2;
}

__global__ void cg_init_kernel(float* __restrict__ cg){
  int idx = blockIdx.x * blockDim.x + threadIdx.x;
  if (idx >= 4096) return;
  int n = idx & 15, m = (idx >> 4) & 15, l = idx >> 8;
  int l1 = degi(l), l2 = degi(m), l3 = degi(n);
  float out = 0.0f;
  if (l3 >= iabs(l1 - l2) && l3 <= l1 + l2){
    int mr1 = l - l1*l1 - l1;
    int mr2 = m - l2*l2 - l2;
    int mr3 = n - l3*l3 - l3;
    UT t1[2], t2[2], t3[2];
    int c1 = urow(l1, mr1, t1);
    int c2 = urow(l2, mr2, t2);
    int c3 = urow(l3, mr3, t3);
    double ar = 0.0, ai = 0.0;
    for (int a = 0; a < c1; ++a)
      for (int b = 0; b < c2; ++b)
        for (int c = 0; c < c3; ++c){
          double cgv = cg_cplx(l1, l2, l3, t1[a].m, t2[b].m, t3[c].m);
          if (cgv == 0.0) continue;
          double r12 = t1[a].re*t2[b].re - t1[a].im*t2[b].im;
          double i12 = t1[a].re*t2[b].im + t1[a].im*t2[b].re;
          double rr = r12*t3[c].re + i12*t3[c].im;   // times conj(t3)
          double ii = i12*t3[c].re - r12*t3[c].im;
          ar += rr * cgv; ai += ii * cgv;
        }
    out = (float)(ar + ai);
  }
  cg[idx] = out;
}

// ---------------------------------------------------------------------------
// CDNA5 async global->LDS copy (ASYNCcnt path), 16 bytes per lane.
// ---------------------------------------------------------------------------
__device__ __forceinline__ void async_copy16(unsigned lds_off, const float* gptr){
  asm volatile("global_load_async_to_lds_b128 %0, %1, off"
               :: "v"(lds_off), "v"((unsigned long long)(size_t)gptr)
               : "memory");
}
__device__ __forceinline__ void wait_async_4(){ asm volatile("s_wait_asynccnt 0x4" ::: "memory"); }
__device__ __forceinline__ void wait_async_0(){ asm volatile("s_wait_asynccnt 0x0" ::: "memory"); }

__device__ __forceinline__ v8f wmma4(v2f a, v2f b, v8f c){
  return __builtin_amdgcn_wmma_f32_16x16x4_f32(false, a, false, b, (short)0, c, false, false);
}

// tiles: rows = k(4)*i(2)*l(16) = 128 rows, row stride 20 floats (16 f + 4 pad,
// 16B-aligned rows for b128 async stores, 20*l mod 64 -> 16 distinct banks).
#define ROWS   128
#define RSTR   20
#define TSZ    (ROWS*RSTR)      // 2560 floats per input per buffer
#define BUFSZ  (2*TSZ)          // 5120 floats per buffer (in1|in2)

__device__ __forceinline__ void issue_chunk(const float* __restrict__ in1,
                                            const float* __restrict__ in2,
                                            float* buf, int t, size_t gb, int k0, int f0){
  float* s1 = buf;
  float* s2 = buf + TSZ;
  #pragma unroll
  for (int rep = 0; rep < 2; ++rep){
    int q   = t + rep * 256;        // 0..511
    int row = q >> 2;               // k*32 + i*16 + l
    int fq  = (q & 3) * 4;
    int l = row & 15, i = (row >> 4) & 1, k = row >> 5;
    size_t g = gb + (size_t)(k0 + k) * 4096 + (size_t)i * 2048 + (size_t)l * 128 + f0 + fq;
    unsigned off = (unsigned)(row * RSTR + fq) * 4u;
    async_copy16((unsigned)(size_t)s1 + off, in1 + g);
    async_copy16((unsigned)(size_t)s2 + off, in2 + g);
  }
}

__global__ __launch_bounds__(256) void ti_main_kernel(
    const float* __restrict__ in1, const float* __restrict__ in2,
    const float* __restrict__ wts, const float* __restrict__ ker,
    const float* __restrict__ cg,  float* __restrict__ out)
{
  __shared__ float smem[2*BUFSZ];   // 40 KB: double-buffered tiles; stage2 M scratch aliases it
  __shared__ float scg[4096];       // 16 KB CG table
  float* Msc = smem;

  const int t    = threadIdx.x;
  const int b    = blockIdx.x >> 3;
  const int f0   = (blockIdx.x & 7) * 16;
  const int lane = t & 31;
  const int w    = t >> 5;          // wave 0..7
  const int half = lane >> 4;
  const int mrow = lane & 15;
  const size_t gb = (size_t)b * 131072;

  // prologue: start chunk 0 into buffer 0
  issue_chunk(in1, in2, smem, t, gb, 0, f0);

  for (int q = t; q < 4096; q += 256) scg[q] = cg[q];

  v8f acc[2][4];
  #pragma unroll
  for (int a = 0; a < 2; ++a)
    #pragma unroll
    for (int p = 0; p < 4; ++p)
      #pragma unroll
      for (int v = 0; v < 8; ++v) acc[a][p][v] = 0.0f;

  #pragma unroll
  for (int kc = 0; kc < 8; ++kc){
    float* c1 = smem + (kc & 1) * BUFSZ;
    float* c2 = c1 + TSZ;
    __syncthreads();                       // everyone done reading the other buffer
    if (kc + 1 < 8){
      issue_chunk(in1, in2, smem + ((kc + 1) & 1) * BUFSZ, t, gb, (kc + 1) * 4, f0);
      wait_async_4();                      // chunk kc's 4 copies (issued earlier) done
    } else {
      wait_async_0();
    }
    __syncthreads();                       // buffer kc visible to all waves

    float2 wv = *(const float2*)(wts + kc * 4 + 2 * half);  // weights for this lane's K rows

    #pragma unroll
    for (int ff = 0; ff < 2; ++ff){
      const int f  = 2 * w + ff;           // 0..15 within tile
      const int ka = 2 * half;             // local K index held in element .x (ISA A layout)
      v2f A0, A1, B0, B1;
      A0.x = c1[(((ka  )*2 + 0)*16 + mrow)*RSTR + f] * wv.x;
      A0.y = c1[(((ka+1)*2 + 0)*16 + mrow)*RSTR + f] * wv.y;
      A1.x = c1[(((ka  )*2 + 1)*16 + mrow)*RSTR + f] * wv.x;
      A1.y = c1[(((ka+1)*2 + 1)*16 + mrow)*RSTR + f] * wv.y;
      B0.x = c2[(((ka  )*2 + 0)*16 + mrow)*RSTR + f];
      B0.y = c2[(((ka+1)*2 + 0)*16 + mrow)*RSTR + f];
      B1.x = c2[(((ka  )*2 + 1)*16 + mrow)*RSTR + f];
      B1.y = c2[(((ka+1)*2 + 1)*16 + mrow)*RSTR + f];
      acc[ff][0] = wmma4(A0, B0, acc[ff][0]);   // (i=0,j=0)
      acc[ff][1] = wmma4(A1, B1, acc[ff][1]);   // (i=1,j=1)
      acc[ff][2] = wmma4(A0, B1, acc[ff][2]);   // (i=0,j=1)
      acc[ff][3] = wmma4(A1, B0, acc[ff][3]);   // (i=1,j=0)
    }
  }

  // ---------------- stage 2: CG x kernel contraction ----------------
  for (int ff = 0; ff < 2; ++ff){
    __syncthreads();
    #pragma unroll
    for (int p = 0; p < 4; ++p)
      #pragma unroll
      for (int v = 0; v < 8; ++v)
        Msc[w*1024 + p*256 + (v + 8*half)*16 + mrow] = acc[ff][p][v];
    __syncthreads();

    const int fsel = t >> 5;          // which wave's f
    const int r    = (t >> 4) & 1;
    const int n    = t & 15;
    const int f    = f0 + 2*fsel + ff;
    const int dn   = degi(n);
    const int pa   = r ? 2 : 0;       // r0: (0,0)+(1,1)   r1: (0,1)+(1,0)
    const int pb   = r ? 3 : 1;
    const int ia = 0,          ja = r ? 1 : 0;
    const int ib = 1,          jb = r ? 0 : 1;

    float kfa[16], kfb[16];
    #pragma unroll
    for (int dl = 0; dl < 4; ++dl)
      #pragma unroll
      for (int dm = 0; dm < 4; ++dm){
        kfa[dl*4+dm] = ker[(size_t)ia*32768 + (size_t)dl*8192 + (size_t)ja*4096 + (size_t)dm*1024 + (size_t)r*512 + (size_t)dn*128 + f];
        kfb[dl*4+dm] = ker[(size_t)ib*32768 + (size_t)dl*8192 + (size_t)jb*4096 + (size_t)dm*1024 + (size_t)r*512 + (size_t)dn*128 + f];
      }

    const float* Ma = &Msc[fsel*1024 + pa*256];
    const float* Mb = &Msc[fsel*1024 + pb*256];
    float o = 0.0f;
    for (int l = 0; l < 16; ++l){
      int dl4 = degi(l) * 4;
      #pragma unroll
      for (int m = 0; m < 16; ++m){
        float c = scg[(l*16 + m)*16 + n];
        float s = fmaf(Ma[l*16+m], kfa[dl4 + degi(m)], Mb[l*16+m] * kfb[dl4 + degi(m)]);
        o = fmaf(c, s, o);
      }
    }
    out[(size_t)b*4096 + (size_t)r*2048 + (size_t)n*128 + f] = o;
  }
}

// ---------------------------------------------------------------------------
extern "C" void kernel_launch(void* const* d_in, const int* in_sizes, int n_in,
                              void* d_out, int out_size, void* d_ws, size_t ws_size,
                              hipStream_t stream){
  (void)in_sizes; (void)n_in; (void)out_size; (void)ws_size;
  const float* in1 = (const float*)d_in[0];
  const float* in2 = (const float*)d_in[1];
  const float* wts = (const float*)d_in[2];
  const float* ker = (const float*)d_in[3];
  float* cg = (float*)d_ws;                       // 4096 floats
  cg_init_kernel<<<16, 256, 0, stream>>>(cg);
  ti_main_kernel<<<2048, 256, 0, stream>>>(in1, in2, wts, ker, cg, (float*)d_out);
}